// GCN_2388001817259
// MI455X (gfx1250) — compile-verified
//
#include <hip/hip_runtime.h>

// ---------------------------------------------------------------------------
// GCN (3x GCNConv + JK-cat + global_add_pool + MLP) for MI455X (gfx1250).
// All 128-wide GEMMs run through V_WMMA_F32_16X16X4_F32 (fp32 in, fp32 acc),
// which matches the fp32 reference exactly; the kernel is bound by the
// E=1.6M edge gather/scatter, not by matrix math.
// A-tile LDS staging uses gfx1250 async loads (GLOBAL_LOAD_ASYNC_TO_LDS_B128
// + s_wait_asynccnt) when available.
// ---------------------------------------------------------------------------

#define HD 128          // hidden width (D == H == 128)
#define SA 132          // LDS row stride for A tile (pad 4 floats: avoids 64-bank conflicts)
#define SB 132          // LDS row stride for B^T tile
#define GEMM_LDS_BYTES ((size_t)(HD * SA + HD * SB) * sizeof(float))

typedef __attribute__((ext_vector_type(2))) float v2f;
typedef __attribute__((ext_vector_type(8))) float v8f;
typedef __attribute__((ext_vector_type(4))) int   v4i;

#if __has_builtin(__builtin_amdgcn_global_load_async_to_lds_b128) && \
    __has_builtin(__builtin_amdgcn_s_wait_asynccnt)
#define HAVE_ASYNC_LDS 1
#define GLOBAL_AS __attribute__((address_space(1)))
#define LDS_AS    __attribute__((address_space(3)))
#endif

// ---------------------------------------------------------------------------
// C[M x 128] = A[M x 128] @ B[128 x 128]   (+ C if flags&1) (+bias) (ReLU if flags&2)
// Block: 256 threads = 8 waves. Each block covers 128 rows; wave w owns rows
// [w*16, w*16+16) of the block tile and the full 128 output columns
// (8 accumulators of 16x16). K loop: 32 steps of WMMA f32 16x16x4.
//
// WMMA f32 VGPR layouts (ISA 7.12.2):
//   A 16x4 : lane l -> row (l&15), K = kk + 2*(l>>4) + {0,1}   -> v2f
//   B 4x16 : lane l -> col (l&15), K = kk + 2*(l>>4) + {0,1}   -> v2f (B^T in LDS)
//   C 16x16: vgpr r, lane l -> row r + 8*(l>>4), col (l&15)
// ---------------------------------------------------------------------------
__global__ __launch_bounds__(256) void gemm128_wmma(
    const float* __restrict__ A, const float* __restrict__ B,
    float* __restrict__ C, const float* __restrict__ bias, int M, int flags)
{
  extern __shared__ float lds[];
  float* ldsA  = lds;             // [128][SA]   A tile, row-major
  float* ldsBT = lds + HD * SA;   // [128][SB]   B tile, transposed (col-major)

  const int tid  = threadIdx.x;
  const int row0 = blockIdx.x * HD;
  const int lane = tid & 31;
  const int wave = tid >> 5;        // 0..7 -> 16-row strip

  // ---- stage A tile -------------------------------------------------------
  const bool fullTile = (row0 + HD) <= M;
#ifdef HAVE_ASYNC_LDS
  if (fullTile) {
    // Async DMA: lane l moves 16B of row r straight into LDS (no VGPR pass).
    for (int r = wave; r < HD; r += 8) {
      const float* gsrc = A + (size_t)(row0 + r) * HD + lane * 4;
      float* ldst = &ldsA[r * SA + lane * 4];
      __builtin_amdgcn_global_load_async_to_lds_b128(
          (GLOBAL_AS v4i*)gsrc, (LDS_AS v4i*)ldst, 0, 0);
    }
  } else
#endif
  {
    for (int idx = tid; idx < HD * (HD / 4); idx += 256) {
      const int r  = idx >> 5;     // tile row
      const int c4 = idx & 31;     // float4 column
      float4 v = make_float4(0.f, 0.f, 0.f, 0.f);
      const int row = row0 + r;
      if (row < M) v = *(const float4*)(A + (size_t)row * HD + c4 * 4);
      float* p = &ldsA[r * SA + c4 * 4];
      p[0] = v.x; p[1] = v.y; p[2] = v.z; p[3] = v.w;
    }
  }
  // ---- stage B transposed (VALU path; overlaps with async A fill) ---------
  for (int idx = tid; idx < HD * (HD / 4); idx += 256) {
    const int k  = idx >> 5;
    const int c4 = idx & 31;
    const float4 v = *(const float4*)(B + (size_t)k * HD + c4 * 4);
    ldsBT[(c4 * 4 + 0) * SB + k] = v.x;
    ldsBT[(c4 * 4 + 1) * SB + k] = v.y;
    ldsBT[(c4 * 4 + 2) * SB + k] = v.z;
    ldsBT[(c4 * 4 + 3) * SB + k] = v.w;
  }
#ifdef HAVE_ASYNC_LDS
  if (fullTile) __builtin_amdgcn_s_wait_asynccnt(0);
#endif
  __syncthreads();

  const int mh = lane & 15;       // row (A) / col (B,C) within 16
  const int hi = lane >> 4;       // 0/1
  const int kb = hi * 2;          // K sub-offset {0,2}

  v8f acc[8];
  if (flags & 1) {
#pragma unroll
    for (int nt = 0; nt < 8; ++nt)
#pragma unroll
      for (int r = 0; r < 8; ++r) {
        const int row = row0 + wave * 16 + r + hi * 8;
        acc[nt][r] = (row < M) ? C[(size_t)row * HD + nt * 16 + mh] : 0.f;
      }
  } else {
#pragma unroll
    for (int nt = 0; nt < 8; ++nt)
#pragma unroll
      for (int r = 0; r < 8; ++r) acc[nt][r] = 0.f;
  }

  const float* aRow = &ldsA[(wave * 16 + mh) * SA];
  for (int kk = 0; kk < HD; kk += 4) {
    v2f a;
    a[0] = aRow[kk + kb];
    a[1] = aRow[kk + kb + 1];
#pragma unroll
    for (int nt = 0; nt < 8; ++nt) {
      const float* bp = &ldsBT[(nt * 16 + mh) * SB + kk + kb];
      v2f b;
      b[0] = bp[0];
      b[1] = bp[1];
      acc[nt] = __builtin_amdgcn_wmma_f32_16x16x4_f32(
          false, a, false, b, (short)0, acc[nt], false, false);
    }
  }

#pragma unroll
  for (int nt = 0; nt < 8; ++nt) {
    const float bv = bias ? bias[nt * 16 + mh] : 0.f;
#pragma unroll
    for (int r = 0; r < 8; ++r) {
      const int row = row0 + wave * 16 + r + hi * 8;
      if (row < M) {
        float v = acc[nt][r] + bv;
        if (flags & 2) v = fmaxf(v, 0.f);
        C[(size_t)row * HD + nt * 16 + mh] = v;
      }
    }
  }
}

// ---------------------------------------------------------------------------
// deg[dst] += 1 per edge
// ---------------------------------------------------------------------------
__global__ __launch_bounds__(256) void degree_k(const int* __restrict__ dst,
                                                float* __restrict__ deg, int E)
{
  const int e = blockIdx.x * 256 + threadIdx.x;
  if (e < E) atomicAdd(&deg[dst[e]], 1.0f);
}

__global__ __launch_bounds__(256) void dinv_k(const float* __restrict__ deg,
                                              float* __restrict__ dinv, int N)
{
  const int i = blockIdx.x * 256 + threadIdx.x;
  if (i < N) dinv[i] = rsqrtf(deg[i] + 1.0f);
}

// ---------------------------------------------------------------------------
// agg[dst] += xw[src] * (dinv[src]*dinv[dst]).  One wave per edge; each lane
// owns one float4 of the 128-float row -> coalesced 512B gather + 4 f32
// global atomics per lane. L2 (192MB) holds the whole 51MB feature matrix.
// ---------------------------------------------------------------------------
__global__ __launch_bounds__(256) void edge_scatter(
    const float* __restrict__ xw, const int* __restrict__ src,
    const int* __restrict__ dst, const float* __restrict__ dinv,
    float* __restrict__ agg, int E)
{
  const int tid = blockIdx.x * 256 + threadIdx.x;
  const int e = tid >> 5;
  const int q = tid & 31;
  if (e >= E) return;
  if (q == 0 && e + 4096 < E) {          // gfx1250 global_prefetch_b8 path
    __builtin_prefetch(&src[e + 4096], 0, 0);
    __builtin_prefetch(&dst[e + 4096], 0, 0);
  }
  const int s = src[e];
  const int d = dst[e];
  const float norm = dinv[s] * dinv[d];
  const float4 v = ((const float4*)(xw + (size_t)s * HD))[q];
  float* ap = agg + (size_t)d * HD + q * 4;
  atomicAdd(ap + 0, v.x * norm);
  atomicAdd(ap + 1, v.y * norm);
  atomicAdd(ap + 2, v.z * norm);
  atomicAdd(ap + 3, v.w * norm);
}

// ---------------------------------------------------------------------------
// h_out = relu(agg + xw * dinv^2 + b)   (self-loop + bias + activation)
// ---------------------------------------------------------------------------
__global__ __launch_bounds__(256) void finalize_k(
    const float* __restrict__ agg, const float* __restrict__ xw,
    const float* __restrict__ dinv, const float* __restrict__ b,
    float* __restrict__ hout, int N)
{
  const int idx = blockIdx.x * 256 + threadIdx.x;
  if (idx >= N * HD) return;
  const int i = idx >> 7;
  const int d = idx & (HD - 1);
  const float di = dinv[i];
  const float v = agg[idx] + xw[idx] * di * di + b[d];
  hout[idx] = fmaxf(v, 0.f);
}

// ---------------------------------------------------------------------------
// pooled[batch[i]] += hjk[i]
// ---------------------------------------------------------------------------
__global__ __launch_bounds__(256) void pool_k(
    const float* __restrict__ hjk, const int* __restrict__ batch,
    float* __restrict__ pooled, int N)
{
  const int idx = blockIdx.x * 256 + threadIdx.x;
  if (idx >= N * HD) return;
  const int i = idx >> 7;
  const int d = idx & (HD - 1);
  atomicAdd(&pooled[(size_t)batch[i] * HD + d], hjk[idx]);
}

// ---------------------------------------------------------------------------
// out[g][o] = t[g] . Wm2[:,o] + bm2[o]     (512 x 10, tiny)
// ---------------------------------------------------------------------------
__global__ __launch_bounds__(256) void head_k(
    const float* __restrict__ t, const float* __restrict__ Wm2,
    const float* __restrict__ bm2, float* __restrict__ out, int G, int OUT)
{
  const int idx = blockIdx.x * 256 + threadIdx.x;
  if (idx >= G * OUT) return;
  const int g = idx / OUT;
  const int o = idx - g * OUT;
  float s = bm2[o];
  for (int d = 0; d < HD; ++d) s += t[(size_t)g * HD + d] * Wm2[d * OUT + o];
  out[idx] = s;
}

// ---------------------------------------------------------------------------
extern "C" void kernel_launch(void* const* d_in, const int* in_sizes, int n_in,
                              void* d_out, int out_size, void* d_ws, size_t ws_size,
                              hipStream_t stream)
{
  (void)n_in; (void)out_size; (void)ws_size;

  const float* x     = (const float*)d_in[0];
  const int*   eidx  = (const int*)d_in[1];
  const int*   batch = (const int*)d_in[2];
  const float* W0  = (const float*)d_in[3];  const float* b0  = (const float*)d_in[4];
  const float* W1  = (const float*)d_in[5];  const float* b1  = (const float*)d_in[6];
  const float* W2  = (const float*)d_in[7];  const float* b2  = (const float*)d_in[8];
  const float* Wjk = (const float*)d_in[9];  const float* bjk = (const float*)d_in[10];
  const float* Wm1 = (const float*)d_in[11]; const float* bm1 = (const float*)d_in[12];
  const float* Wm2 = (const float*)d_in[13]; const float* bm2 = (const float*)d_in[14];

  const int N = in_sizes[2];        // batch vector has one entry per node
  const int E = in_sizes[1] / 2;    // edge_index is [2, E]
  const int G = 512, OUT = 10;
  const int* src = eidx;
  const int* dst = eidx + E;

  // workspace layout (fp32)
  float* p = (float*)d_ws;
  float* deg    = p; p += N;
  float* dinv   = p; p += N;
  float* xw     = p; p += (size_t)N * HD;
  float* agg    = p; p += (size_t)N * HD;
  float* h1     = p; p += (size_t)N * HD;
  float* h2     = p; p += (size_t)N * HD;
  float* h3     = p; p += (size_t)N * HD;
  float* pooled = p; p += (size_t)G * HD;
  float* tbuf   = p; p += (size_t)G * HD;
  float* hjk    = xw;               // xw buffer is free after layer 3 finalize

  (void)hipFuncSetAttribute((const void*)gemm128_wmma,
                            hipFuncAttributeMaxDynamicSharedMemorySize,
                            (int)GEMM_LDS_BYTES);

  // degrees + 1/sqrt(deg)
  (void)hipMemsetAsync(deg, 0, (size_t)N * sizeof(float), stream);
  degree_k<<<(E + 255) / 256, 256, 0, stream>>>(dst, deg, E);
  dinv_k<<<(N + 255) / 256, 256, 0, stream>>>(deg, dinv, N);

  const float* Ws[3]    = {W0, W1, W2};
  const float* bs[3]    = {b0, b1, b2};
  float*       houts[3] = {h1, h2, h3};
  const float* hin = x;

  const int gemmGrid = (N + HD - 1) / HD;
  const int elemGrid = ((N * HD) + 255) / 256;
  const long long sc = (long long)E * 32;
  const int scatGrid = (int)((sc + 255) / 256);

  for (int l = 0; l < 3; ++l) {
    gemm128_wmma<<<gemmGrid, 256, GEMM_LDS_BYTES, stream>>>(hin, Ws[l], xw, nullptr, N, 0);
    (void)hipMemsetAsync(agg, 0, (size_t)N * HD * sizeof(float), stream);
    edge_scatter<<<scatGrid, 256, 0, stream>>>(xw, src, dst, dinv, agg, E);
    finalize_k<<<elemGrid, 256, 0, stream>>>(agg, xw, dinv, bs[l], houts[l], N);
    hin = houts[l];
  }

  // JK 'cat' projection without materializing the concat:
  // hjk = h1@Wjk[0:128] + h2@Wjk[128:256] + h3@Wjk[256:384] + bjk
  gemm128_wmma<<<gemmGrid, 256, GEMM_LDS_BYTES, stream>>>(h1, Wjk,               hjk, nullptr, N, 0);
  gemm128_wmma<<<gemmGrid, 256, GEMM_LDS_BYTES, stream>>>(h2, Wjk + HD * HD,     hjk, nullptr, N, 1);
  gemm128_wmma<<<gemmGrid, 256, GEMM_LDS_BYTES, stream>>>(h3, Wjk + 2 * HD * HD, hjk, bjk,     N, 1);

  // global_add_pool
  (void)hipMemsetAsync(pooled, 0, (size_t)G * HD * sizeof(float), stream);
  pool_k<<<elemGrid, 256, 0, stream>>>(hjk, batch, pooled, N);

  // MLP head: relu(pooled@Wm1+bm1) @ Wm2 + bm2
  gemm128_wmma<<<(G + HD - 1) / HD, 256, GEMM_LDS_BYTES, stream>>>(pooled, Wm1, tbuf, bm1, G, 2);
  head_k<<<(G * OUT + 255) / 256, 256, 0, stream>>>(tbuf, Wm2, bm2, (float*)d_out, G, OUT);
}